// SparseConv3d_82429012345627
// MI455X (gfx1250) — compile-verified
//
#include <hip/hip_runtime.h>

typedef float v2f __attribute__((ext_vector_type(2)));
typedef float v8f __attribute__((ext_vector_type(8)));

#define NPTS   400000
#define CIN    32
#define COUT   64
#define KVOL   27
#define WSLICE (CIN * COUT)          // 2048 floats per offset slice
#define LSTRIDE 34                   // padded ci-stride (even -> 8B-aligned b64 LDS loads)
#define LSLICE  (COUT * LSTRIDE)     // 2176 floats per LDS slice

// Zero one feature row (the reference's feats_ext sentinel row) into d_ws.
__global__ void ZeroPage_kernel(float* zrow) {
    if (threadIdx.x < CIN) zrow[threadIdx.x] = 0.0f;
}

// D = sum_k gather(feats, nbr[:,k]) @ W[k] + bias, via V_WMMA_F32_16X16X4_F32.
// One wave = 16 rows x 64 cols output tile.
__global__ __launch_bounds__(256) void SparseConv3d_wmma_kernel(
    const float* __restrict__ feats,   // [N][32]
    const float* __restrict__ weight,  // [27][32][64]  (ci-major, co contiguous)
    const float* __restrict__ bias,    // [64]
    const int*   __restrict__ nbr,     // [N][27], sentinel == N
    const float* __restrict__ zrow,    // [32] zeros (feats_ext sentinel row)
    float* __restrict__ out)           // [N][64]
{
    __shared__ float lds[2 * LSLICE]; // double-buffered transposed weight slice [co][ci]

    const int tid   = threadIdx.x;
    const int lane  = tid & 31;
    const int wave  = tid >> 5;
    const int row16 = lane & 15;      // M index (A rows / D cols within tile)
    const int grp   = lane >> 4;      // K-group: 0 -> K pair (0,1), 1 -> (2,3)

    const int ntiles = NPTS / 16;     // 25000, divides exactly
    int tile = blockIdx.x * 8 + wave;
    if (tile >= ntiles) tile = ntiles - 1;   // (defensive; grid is exact)
    const int base = tile * 16;
    const int row  = base + row16;

    v8f acc[4] = {};

    // Preload transposed slice for k = 0
    {
        const float* wk = weight;
        #pragma unroll
        for (int i = 0; i < 8; ++i) {
            int e = tid + i * 256;                       // e = ci*64 + co
            lds[(e & 63) * LSTRIDE + (e >> 6)] = wk[e];  // lds[co][ci]
        }
    }
    __syncthreads();

    for (int k = 0; k < KVOL; ++k) {
        float* cur = &lds[(k & 1) * LSLICE];

        // Prefetch next weight slice (transposed) into the other LDS buffer;
        // its global-load latency hides under this iteration's 32 WMMAs.
        if (k + 1 < KVOL) {
            float* nxt = &lds[((k + 1) & 1) * LSLICE];
            const float* wk = weight + (size_t)(k + 1) * WSLICE;
            #pragma unroll
            for (int i = 0; i < 8; ++i) {
                int e = tid + i * 256;
                nxt[(e & 63) * LSTRIDE + (e >> 6)] = wk[e];
            }
        }

        // Rulebook gather: per-lane row pointer for this offset (lanes 16-31
        // duplicate lanes 0-15; both K-groups need the same 16 rows).
        // Sentinel rows read the zero page -> no mask VALU in the WMMA chain.
        const int idx = nbr[(size_t)row * KVOL + k];
        const float* arow = (idx < NPTS) ? (feats + (size_t)idx * CIN) : zrow;

        #pragma unroll
        for (int s = 0; s < 8; ++s) {                     // K = 32 in steps of 4
            const int c0 = s * 4 + grp * 2;
            v2f a = *(const v2f*)(arow + c0);             // global b64 gather
            #pragma unroll
            for (int t = 0; t < 4; ++t) {                 // 4 col-tiles of 16
                const int co = t * 16 + row16;
                v2f b = *(const v2f*)(cur + co * LSTRIDE + c0);  // ds_load_b64
                acc[t] = __builtin_amdgcn_wmma_f32_16x16x4_f32(
                    false, a, false, b, (short)0, acc[t], false, false);
            }
        }
        __syncthreads();   // reads of cur done before it becomes nxt; nxt ready
    }

    // Writeback D fragments + bias.  VGPR j holds row (j + 8*grp), col = t*16+row16.
    #pragma unroll
    for (int t = 0; t < 4; ++t) {
        const int co = t * 16 + row16;
        const float bv = bias[co];
        #pragma unroll
        for (int j = 0; j < 8; ++j) {
            const int r = base + j + grp * 8;
            out[(size_t)r * COUT + co] = acc[t][j] + bv;
        }
    }
}

extern "C" void kernel_launch(void* const* d_in, const int* in_sizes, int n_in,
                              void* d_out, int out_size, void* d_ws, size_t ws_size,
                              hipStream_t stream) {
    const float* feats  = (const float*)d_in[0];
    const float* weight = (const float*)d_in[1];
    const float* bias   = (const float*)d_in[2];
    const int*   nbr    = (const int*)d_in[3];
    float*       out    = (float*)d_out;
    float*       zrow   = (float*)d_ws;      // 32-float zero page

    hipLaunchKernelGGL(ZeroPage_kernel, dim3(1), dim3(32), 0, stream, zrow);

    const int ntiles = NPTS / 16;            // 25000
    const int blocks = (ntiles + 7) / 8;     // 3125 (exact: 8 waves/block)
    hipLaunchKernelGGL(SparseConv3d_wmma_kernel, dim3(blocks), dim3(256), 0, stream,
                       feats, weight, bias, nbr, zrow, out);
}